// LoLa_48284022342007
// MI455X (gfx1250) — compile-verified
//
#include <hip/hip_runtime.h>

// CDNA5 / gfx1250, wave32. f32 WMMA path (V_WMMA_F32_16X16X4_F32).
// out(131072 x 1025) = [rowsum(x) | x | x @ w], fused single kernel.

typedef __attribute__((ext_vector_type(2))) float v2f;
typedef __attribute__((ext_vector_type(4))) float v4f;
typedef __attribute__((ext_vector_type(8))) float v8f;

constexpr int KP         = 512;            // inner dim P
constexpr int NOUT       = 512;            // w columns
constexpr int QOUT       = 1 + KP + NOUT;  // 1025 output columns
constexpr int MTILE      = 64;             // rows per block
constexpr int LDS_STRIDE = KP + 4;         // +4 floats: stride%64banks==4 -> conflict-free A frags

__global__ __launch_bounds__(256) void lola_fused(const float* __restrict__ x,
                                                  const float* __restrict__ w,
                                                  float* __restrict__ out) {
    __shared__ float xs[MTILE * LDS_STRIDE];   // ~129 KB of 320 KB WGP LDS
    __shared__ float rowsum[MTILE];

    const int tid  = threadIdx.x;
    const int row0 = blockIdx.x * MTILE;

    if (tid < MTILE) rowsum[tid] = 0.0f;
    __syncthreads();

    // ---- Stage x tile -> LDS; fuse identity-copy stores + row-sum ----
    // thread t owns column-quad (t & 127), rows (t>>7) + 2j (coalesced 16B loads).
    // x is read-once and out is write-once -> non-temporal, keep L2 for w.
    {
        const int c4 = (tid & 127) * 4;
        for (int r = (tid >> 7); r < MTILE; r += 2) {
            const float* src = x + (size_t)(row0 + r) * KP + c4;
            v4f v = __builtin_nontemporal_load((const v4f*)src);
            *(v4f*)&xs[r * LDS_STRIDE + c4] = v;
            // out[., 1+p]: row base = r*1025+1 -> only 4B aligned, scalar stores
            float* dst = out + (size_t)(row0 + r) * QOUT + 1 + c4;
            __builtin_nontemporal_store(v.x, dst + 0);
            __builtin_nontemporal_store(v.y, dst + 1);
            __builtin_nontemporal_store(v.z, dst + 2);
            __builtin_nontemporal_store(v.w, dst + 3);
            atomicAdd(&rowsum[r], (v.x + v.y) + (v.z + v.w));   // ds_add_f32
        }
    }
    __syncthreads();
    if (tid < MTILE)
        __builtin_nontemporal_store(rowsum[tid], out + (size_t)(row0 + tid) * QOUT);

    // ---- GEMM: out[., 513+q] = x_tile @ w  via V_WMMA_F32_16X16X4_F32 ----
    // Wave arrangement 1(M) x 8(N): each wave owns all 4 M-subtiles (A frags
    // from LDS are cheap) and a unique 64-col N strip -> each B fragment
    // (2 VMEM loads) feeds 4 WMMAs; zero cross-wave B redundancy.
    const int wave = tid >> 5;      // 8 waves (wave32), N strip = wave*64
    const int lane = tid & 31;
    const int half = lane >> 4;     // K sub-phase per ISA A/B layout
    const int l    = lane & 15;

    v8f acc[4][4];                  // [mtile][ntile], 128 VGPRs
#pragma unroll
    for (int mt = 0; mt < 4; ++mt)
#pragma unroll
        for (int nt = 0; nt < 4; ++nt) acc[mt][nt] = v8f{};

    const float* wcol = w + (size_t)(wave * 64 + l);

    for (int k = 0; k < KP; k += 4) {
        // A frags (16x4 f32): lanes 0-15 -> K=k,k+1 ; lanes 16-31 -> K=k+2,k+3
        v2f a[4];
#pragma unroll
        for (int mt = 0; mt < 4; ++mt)
            a[mt] = *(const v2f*)&xs[(mt * 16 + l) * LDS_STRIDE + k + 2 * half];

        const float* wb = wcol + (size_t)(k + 2 * half) * NOUT;
#pragma unroll
        for (int nt = 0; nt < 4; ++nt) {
            // B frag (4x16 f32): VGPR0 = row k (k+2 hi half), VGPR1 = row k+1 (k+3)
            v2f b;
            b.x = wb[nt * 16];
            b.y = wb[nt * 16 + NOUT];
#pragma unroll
            for (int mt = 0; mt < 4; ++mt)
                // (neg_a, A, neg_b, B, c_mod, C, reuse_a, reuse_b)
                acc[mt][nt] = __builtin_amdgcn_wmma_f32_16x16x4_f32(
                    false, a[mt], false, b, (short)0, acc[mt][nt], false, false);
        }
    }

    // ---- Store D: VGPR j -> row M=j (lanes 0-15) / M=8+j (lanes 16-31) ----
#pragma unroll
    for (int mt = 0; mt < 4; ++mt) {
        const size_t rb = (size_t)(row0 + mt * 16 + half * 8);
#pragma unroll
        for (int nt = 0; nt < 4; ++nt) {
            const int col = 513 + wave * 64 + nt * 16 + l;
#pragma unroll
            for (int j = 0; j < 8; ++j)
                __builtin_nontemporal_store(acc[mt][nt][j],
                                            &out[(rb + j) * QOUT + col]);
        }
    }
}

extern "C" void kernel_launch(void* const* d_in, const int* in_sizes, int n_in,
                              void* d_out, int out_size, void* d_ws, size_t ws_size,
                              hipStream_t stream) {
    const float* x  = (const float*)d_in[0];   // (32768, 4, 512) f32
    const float* w  = (const float*)d_in[1];   // (512, 512) f32
    float* out      = (float*)d_out;           // (32768, 4, 1025) f32

    const int rows = 32768 * 4;                // 131072, divisible by MTILE
    hipLaunchKernelGGL(lola_fused, dim3(rows / MTILE), dim3(256), 0, stream,
                       x, w, out);
}